// Decoder_82111184765559
// MI455X (gfx1250) — compile-verified
//
#include <hip/hip_runtime.h>
#include <hip/hip_bf16.h>

typedef float v2f __attribute__((ext_vector_type(2)));
typedef float v8f __attribute__((ext_vector_type(8)));

#define N_LSTM 256
#define ZDIM   1024   // 4 * N_LSTM
#define FEAT   5
#define ROWS_PER_WG 32
#define HSTRIDE 258   // padded LDS row stride (even -> 8B-aligned b64 A reads)

__global__ __launch_bounds__(256)
void lstm_step_wmma_f32(const float* __restrict__ x,      // (B,5)
                        const float* __restrict__ h0,     // (B,256)
                        const float* __restrict__ c0,     // (B,256)
                        const float* __restrict__ W,      // (5,1024)
                        const float* __restrict__ U,      // (256,1024)
                        const float* __restrict__ b,      // (1024)
                        const float* __restrict__ Wd,     // (256,5)
                        const float* __restrict__ bd,     // (5)
                        float* __restrict__ logits_out,   // (B,5)
                        float* __restrict__ h_out,        // (B,256)
                        float* __restrict__ c_out)        // (B,256)
{
  __shared__ float lds_h[ROWS_PER_WG * HSTRIDE];
  __shared__ float lds_x[ROWS_PER_WG * FEAT];

  const int tid   = threadIdx.x;
  const int wave  = tid >> 5;
  const int lane  = tid & 31;
  const int lhalf = lane >> 4;   // 0: lanes 0-15, 1: lanes 16-31
  const int lmod  = lane & 15;
  const int row0  = blockIdx.x * ROWS_PER_WG;

  // ---- Stage h0 tile (32 x 256 f32) into LDS, padded stride ----
  for (int i = tid; i < ROWS_PER_WG * (N_LSTM / 4); i += 256) {
    const int r  = i >> 6;            // 64 float4 per row
    const int c4 = (i & 63) << 2;
    const float4 v = *(const float4*)(h0 + (size_t)(row0 + r) * N_LSTM + c4);
    float* dst = lds_h + r * HSTRIDE + c4;
    dst[0] = v.x; dst[1] = v.y; dst[2] = v.z; dst[3] = v.w;
  }
  if (tid < ROWS_PER_WG * FEAT)
    lds_x[tid] = x[(size_t)row0 * FEAT + tid];
  __syncthreads();

  // Accumulators: [gate][col-tile][row-tile], each 16x16 f32 C/D = v8f.
  // C/D layout: VGPR r, lane l -> M = r + 8*(l>>4), N = l&15.
  v8f acc[4][2][2];

  // ---- Init acc = b + x @ W (K=5, folded per-lane) ----
#pragma unroll
  for (int g = 0; g < 4; ++g) {
#pragma unroll
    for (int t = 0; t < 2; ++t) {
      const int n = g * 256 + wave * 32 + t * 16 + lmod;   // column in z
      const float w0 = W[0 * ZDIM + n];
      const float w1 = W[1 * ZDIM + n];
      const float w2 = W[2 * ZDIM + n];
      const float w3 = W[3 * ZDIM + n];
      const float w4 = W[4 * ZDIM + n];
      const float bb = b[n];
#pragma unroll
      for (int rt = 0; rt < 2; ++rt) {
#pragma unroll
        for (int r = 0; r < 8; ++r) {
          const int m = rt * 16 + r + 8 * lhalf;           // row within tile
          float s = bb;
          s += lds_x[m * FEAT + 0] * w0;
          s += lds_x[m * FEAT + 1] * w1;
          s += lds_x[m * FEAT + 2] * w2;
          s += lds_x[m * FEAT + 3] * w3;
          s += lds_x[m * FEAT + 4] * w4;
          acc[g][t][rt][r] = s;
        }
      }
    }
  }

  // B-matrix column base pointers: lane l holds column n, rows split by lane half.
  const float* ub[4][2];
#pragma unroll
  for (int g = 0; g < 4; ++g)
#pragma unroll
    for (int t = 0; t < 2; ++t)
      ub[g][t] = U + (size_t)(2 * lhalf) * ZDIM + (g * 256 + wave * 32 + t * 16 + lmod);

  // ---- Main GEMM loop: z += h0_tile @ U, K = 256 in steps of 4 ----
#pragma unroll 2
  for (int k = 0; k < N_LSTM; k += 4) {
    // A fragments (16x4 f32): VGPR0 = h[m][k + 2*lhalf], VGPR1 = h[m][k+1 + 2*lhalf]
    v2f afrag[2];
#pragma unroll
    for (int rt = 0; rt < 2; ++rt) {
      const int off = (rt * 16 + lmod) * HSTRIDE + k + 2 * lhalf;
      v2f a;
      a.x = lds_h[off];
      a.y = lds_h[off + 1];
      afrag[rt] = a;
    }
#pragma unroll
    for (int g = 0; g < 4; ++g) {
#pragma unroll
      for (int t = 0; t < 2; ++t) {
        // B fragment (4x16 f32): VGPR0 = U[k + 2*lhalf][n], VGPR1 = U[k+1+2*lhalf][n]
        v2f bf;
        bf.x = ub[g][t][(size_t)k * ZDIM];
        bf.y = ub[g][t][(size_t)(k + 1) * ZDIM];
        acc[g][t][0] = __builtin_amdgcn_wmma_f32_16x16x4_f32(
            false, afrag[0], false, bf, (short)0, acc[g][t][0], false, false);
        acc[g][t][1] = __builtin_amdgcn_wmma_f32_16x16x4_f32(
            false, afrag[1], false, bf, (short)0, acc[g][t][1], false, false);
      }
    }
  }

  __syncthreads();   // done reading h0 tile from LDS; about to overwrite with h

  // ---- Fused gate epilogue: c = f*c0 + i*relu(zc); h = o*relu(c) ----
#pragma unroll
  for (int t = 0; t < 2; ++t) {
#pragma unroll
    for (int rt = 0; rt < 2; ++rt) {
#pragma unroll
      for (int r = 0; r < 8; ++r) {
        const int m    = rt * 16 + r + 8 * lhalf;
        const int row  = row0 + m;
        const int jcol = wave * 32 + t * 16 + lmod;
        const float zi = acc[0][t][rt][r];
        const float zf = acc[1][t][rt][r];
        const float zc = acc[2][t][rt][r];
        const float zo = acc[3][t][rt][r];
        const float ig = 1.0f / (1.0f + __expf(-zi));
        const float fg = 1.0f / (1.0f + __expf(-zf));
        const float og = 1.0f / (1.0f + __expf(-zo));
        const float c0v = c0[(size_t)row * N_LSTM + jcol];
        const float cc  = fg * c0v + ig * fmaxf(zc, 0.0f);
        const float hh  = og * fmaxf(cc, 0.0f);
        c_out[(size_t)row * N_LSTM + jcol] = cc;
        h_out[(size_t)row * N_LSTM + jcol] = hh;
        lds_h[m * HSTRIDE + jcol] = hh;   // stage h for logits reduction
      }
    }
  }

  __syncthreads();

  // ---- logits = h @ Wd + bd  (32 rows x 5 cols per workgroup) ----
  if (tid < ROWS_PER_WG * FEAT) {
    const int rrow = tid / FEAT;
    const int p    = tid % FEAT;
    float s = bd[p];
    const float* hr = lds_h + rrow * HSTRIDE;
#pragma unroll 4
    for (int j = 0; j < N_LSTM; ++j)
      s += hr[j] * Wd[j * FEAT + p];
    logits_out[(size_t)(row0 + rrow) * FEAT + p] = s;
  }
}

extern "C" void kernel_launch(void* const* d_in, const int* in_sizes, int n_in,
                              void* d_out, int out_size, void* d_ws, size_t ws_size,
                              hipStream_t stream) {
  (void)in_sizes; (void)n_in; (void)d_ws; (void)ws_size; (void)out_size;
  const float* x  = (const float*)d_in[0];   // seq_in (B,1,5)
  const float* h0 = (const float*)d_in[1];
  const float* c0 = (const float*)d_in[2];
  const float* W  = (const float*)d_in[3];
  const float* U  = (const float*)d_in[4];
  const float* b  = (const float*)d_in[5];
  const float* Wd = (const float*)d_in[6];
  const float* bd = (const float*)d_in[7];

  const int BATCH = 65536;
  float* logits = (float*)d_out;                         // (B,5)
  float* h_out  = logits + (size_t)BATCH * FEAT;         // (B,256)
  float* c_out  = h_out + (size_t)BATCH * N_LSTM;        // (B,256)

  dim3 grid(BATCH / ROWS_PER_WG);   // 2048
  dim3 block(256);                  // 8 wave32
  lstm_step_wmma_f32<<<grid, block, 0, stream>>>(x, h0, c0, W, U, b, Wd, bd,
                                                 logits, h_out, c_out);
}